// HNN_structure_cart_embed_3607772529331
// MI455X (gfx1250) — compile-verified
//
#include <hip/hip_runtime.h>

#define HID 256

typedef __attribute__((ext_vector_type(2))) float v2f;
typedef __attribute__((ext_vector_type(8))) float v8f;

// D = A(16x4 f32) x B(4x16 f32) + C(16x16 f32), wave32 WMMA
__device__ inline v8f wmma_k4(float a0, float a1, float b0, float b1, v8f c) {
    v2f A; A.x = a0; A.y = a1;
    v2f B; B.x = b0; B.y = b1;
    return __builtin_amdgcn_wmma_f32_16x16x4_f32(false, A, false, B, (short)0, c, false, false);
}

// tanh via hardware v_exp_f32 + v_rcp_f32 (avoids IEEE div expansion)
__device__ inline float fast_tanh(float x) {
    float e = __expf(2.0f * x);                     // v_exp_f32
    float r = __builtin_amdgcn_rcpf(e + 1.0f);      // v_rcp_f32
    return __builtin_fmaf(-2.0f, r, 1.0f);          // 1 - 2/(e+1)
}

// wave-internal LDS ordering: stores above must complete before loads below
__device__ inline void lds_fence() {
    asm volatile("s_wait_dscnt 0x0" ::: "memory");
}

template <int N>
__device__ inline void xor_reduce16(float* v) {
#pragma unroll
    for (int off = 1; off < 16; off <<= 1) {
#pragma unroll
        for (int i = 0; i < N; i++) v[i] += __shfl_xor(v[i], off, 32);
    }
}

__global__ __launch_bounds__(256) void hnn_fused_kernel(
    const float* __restrict__ y, float* __restrict__ out,
    const float* __restrict__ MW1, const float* __restrict__ Mb1,
    const float* __restrict__ MW2, const float* __restrict__ Mb2,
    const float* __restrict__ HW1, const float* __restrict__ Hb1,
    const float* __restrict__ HW2,
    const float* __restrict__ GW1, const float* __restrict__ Gb1,
    const float* __restrict__ GW2, const float* __restrict__ Gb2)
{
    __shared__ __attribute__((aligned(16))) float sMW1[3 * HID];
    __shared__ __attribute__((aligned(16))) float sMb1[HID];
    __shared__ __attribute__((aligned(16))) float sMW2[HID * 4];
    __shared__ __attribute__((aligned(16))) float sHW1[5 * HID];
    __shared__ __attribute__((aligned(16))) float sHb1[HID];
    __shared__ __attribute__((aligned(16))) float sHW2[HID];
    __shared__ __attribute__((aligned(16))) float sGW1[3 * HID];
    __shared__ __attribute__((aligned(16))) float sGb1[HID];
    __shared__ __attribute__((aligned(16))) float sGW2[HID * 2];
    __shared__ __attribute__((aligned(16))) float sStage[8 * 16 * 4];

    const int tid = threadIdx.x;
    // cooperative weight staging (weights are tiny; one pass, then LDS-resident)
    for (int i = tid; i < 3 * HID; i += 256) sMW1[i] = MW1[i];
    for (int i = tid; i < HID;     i += 256) sMb1[i] = Mb1[i];
    for (int i = tid; i < 4 * HID; i += 256) sMW2[i] = MW2[i];
    for (int i = tid; i < 5 * HID; i += 256) sHW1[i] = HW1[i];
    for (int i = tid; i < HID;     i += 256) sHb1[i] = Hb1[i];
    for (int i = tid; i < HID;     i += 256) sHW2[i] = HW2[i];
    for (int i = tid; i < 3 * HID; i += 256) sGW1[i] = GW1[i];
    for (int i = tid; i < HID;     i += 256) sGb1[i] = Gb1[i];
    for (int i = tid; i < 2 * HID; i += 256) sGW2[i] = GW2[i];
    __syncthreads();

    const int wave = tid >> 5;
    const int l    = tid & 31;
    const int lo   = l & 15;   // A-layout: row index; C-layout: N index
    const int hi   = l >> 4;   // lane half: K split (A/B), M+8 (C)
    float* stg = &sStage[wave * 64];

    const int tile    = blockIdx.x * 8 + wave;
    const int rowBase = tile * 16;

    // this lane's A-row of y (row = rowBase+lo), 24B stride, 8B aligned
    const float* yr = y + (size_t)(rowBase + lo) * 6;
    const float y0 = yr[0], y1 = yr[1], y2 = yr[2];

    // A-operand for x3 (K=3, pad K=3 with 0): VGPR v holds K = v + 2*hi
    const float ax0 = hi ? y2 : y0;
    const float ax1 = hi ? 0.0f : y1;

    // ---------------- Phase 1: M-net fwd + g-net (share A) ----------------
    float aac[32];  // a accum: 8 rows x 4 outs
    float gac[16];  // g accum: 8 rows x 2 outs
#pragma unroll
    for (int i = 0; i < 32; i++) aac[i] = 0.0f;
#pragma unroll
    for (int i = 0; i < 16; i++) gac[i] = 0.0f;

#pragma unroll 2
    for (int nt = 0; nt < 16; nt++) {
        const int col = nt * 16 + lo;
        // B tile of M_W1 (rows = K): VGPR v holds K = v + 2*hi; K=3 padded 0
        const float bm0 = sMW1[(hi ? 2 : 0) * HID + col];
        const float bm1 = hi ? 0.0f : sMW1[HID + col];
        v8f c = {};
        c = wmma_k4(ax0, ax1, bm0, bm1, c);
        const float bias = sMb1[col];
        const float4 w4 = *(const float4*)&sMW2[col * 4];
#pragma unroll
        for (int rr = 0; rr < 8; rr++) {
            const float t = fast_tanh(c[rr] + bias);
            aac[rr * 4 + 0] += t * w4.x;
            aac[rr * 4 + 1] += t * w4.y;
            aac[rr * 4 + 2] += t * w4.z;
            aac[rr * 4 + 3] += t * w4.w;
        }
        const float bg0 = sGW1[(hi ? 2 : 0) * HID + col];
        const float bg1 = hi ? 0.0f : sGW1[HID + col];
        v8f cg = {};
        cg = wmma_k4(ax0, ax1, bg0, bg1, cg);
        const float biasg = sGb1[col];
        const float2 wg = *(const float2*)&sGW2[col * 2];
#pragma unroll
        for (int rr = 0; rr < 8; rr++) {
            const float t = fast_tanh(cg[rr] + biasg);
            gac[rr * 2 + 0] += t * wg.x;
            gac[rr * 2 + 1] += t * wg.y;
        }
    }
    xor_reduce16<32>(aac);
    xor_reduce16<16>(gac);

    // per-row 2x2 algebra (rows rr + 8*hi, replicated across the 16-lane group)
    float p0a[8], p1a[8], m00a[8], m01a[8], m11a[8];
    const float mb2_0 = Mb2[0], mb2_1 = Mb2[1], mb2_2 = Mb2[2], mb2_3 = Mb2[3];
#pragma unroll
    for (int rr = 0; rr < 8; rr++) {
        const float a0 = aac[rr * 4 + 0] + mb2_0;
        const float a1 = aac[rr * 4 + 1] + mb2_1;
        const float a2 = aac[rr * 4 + 2] + mb2_2;
        const float a3 = aac[rr * 4 + 3] + mb2_3;
        aac[rr * 4 + 0] = a0; aac[rr * 4 + 1] = a1;
        aac[rr * 4 + 2] = a2; aac[rr * 4 + 3] = a3;
        const float m00 = a0 * a0 + a1 * a1 + 1.0f;
        const float m01 = a0 * a2 + a1 * a3;
        const float m11 = a2 * a2 + a3 * a3 + 1.0f;
        const int gr = rowBase + rr + 8 * hi;
        const float* yw = y + (size_t)gr * 6;
        const float xd0 = yw[3], xd1 = yw[4];
        const float inv = 1.0f / (m00 * m11 - m01 * m01);  // precise (solve)
        const float p0 = (m11 * xd0 - m01 * xd1) * inv;
        const float p1 = (m00 * xd1 - m01 * xd0) * inv;
        p0a[rr] = p0; p1a[rr] = p1;
        m00a[rr] = m00; m01a[rr] = m01; m11a[rr] = m11;
        if (lo == 0) { stg[(rr + 8 * hi) * 4 + 0] = p0; stg[(rr + 8 * hi) * 4 + 1] = p1; }
    }
    lds_fence();
    const float p0l = stg[lo * 4 + 0];
    const float p1l = stg[lo * 4 + 1];

    // ---------------- Phase 2: H-net value + input-gradient ----------------
    // xp = [x0,x1,x2,p0,p1]; K=5 -> two K-steps of 4
    const float ah0_0 = hi ? y2  : y0;   // K = 0/2
    const float ah1_0 = hi ? p0l : y1;   // K = 1/3
    const float ah0_1 = hi ? 0.0f : p1l; // K = 4/6
    float dhac[40];  // 8 rows x dH[0..4]
#pragma unroll
    for (int i = 0; i < 40; i++) dhac[i] = 0.0f;

#pragma unroll 2
    for (int nt = 0; nt < 16; nt++) {
        const int col = nt * 16 + lo;
        const float b00 = sHW1[(2 * hi) * HID + col];
        const float b10 = sHW1[(2 * hi + 1) * HID + col];
        const float b01 = hi ? 0.0f : sHW1[4 * HID + col];
        v8f c = {};
        c = wmma_k4(ah0_0, ah1_0, b00, b10, c);
        c = wmma_k4(ah0_1, 0.0f, b01, 0.0f, c);
        const float bias = sHb1[col];
        const float w2h  = sHW2[col];
        const float wj0 = sHW1[col];
        const float wj1 = sHW1[HID + col];
        const float wj2 = sHW1[2 * HID + col];
        const float wj3 = sHW1[3 * HID + col];
        const float wj4 = sHW1[4 * HID + col];
#pragma unroll
        for (int rr = 0; rr < 8; rr++) {
            const float t  = fast_tanh(c[rr] + bias);
            const float rv = (1.0f - t * t) * w2h;
            dhac[rr * 5 + 0] += rv * wj0;
            dhac[rr * 5 + 1] += rv * wj1;
            dhac[rr * 5 + 2] += rv * wj2;
            dhac[rr * 5 + 3] += rv * wj3;
            dhac[rr * 5 + 4] += rv * wj4;
        }
    }
    xor_reduce16<40>(dhac);

    float dxa[8], vs1a[8], vs2a[8], dp0a[8], dp1a[8];
    const float gb2_0 = Gb2[0], gb2_1 = Gb2[1];
#pragma unroll
    for (int rr = 0; rr < 8; rr++) {
        const int gr = rowBase + rr + 8 * hi;
        const float* yw = y + (size_t)gr * 6;
        const float cq = yw[1], sq = yw[2], u = yw[5];
        const float dH0 = dhac[rr * 5 + 0];
        const float dH1 = dhac[rr * 5 + 1];
        const float dH2 = dhac[rr * 5 + 2];
        const float dx  = dhac[rr * 5 + 3];
        const float dq  = dhac[rr * 5 + 4];
        const float vs1 = -sq * dq;
        const float vs2 =  cq * dq;
        const float F0 = (gac[rr * 2 + 0] + gb2_0) * u;
        const float F1 = (gac[rr * 2 + 1] + gb2_1) * u;
        dxa[rr] = dx; vs1a[rr] = vs1; vs2a[rr] = vs2;
        dp0a[rr] = -dH0 + F0;
        dp1a[rr] = sq * dH1 - cq * dH2 + F1;
        if (lo == 0) {
            stg[(rr + 8 * hi) * 4 + 0] = dx;
            stg[(rr + 8 * hi) * 4 + 1] = vs1;
            stg[(rr + 8 * hi) * 4 + 2] = vs2;
        }
    }
    lds_fence();
    const float v0l = stg[lo * 4 + 0];
    const float v1l = stg[lo * 4 + 1];
    const float v2l = stg[lo * 4 + 2];
    const float av0 = hi ? v2l : v0l;
    const float av1 = hi ? 0.0f : v1l;

    // ---------------- Phase 3: M-net JVP (recompute tanh state) ----------------
    float dac[32];
#pragma unroll
    for (int i = 0; i < 32; i++) dac[i] = 0.0f;

#pragma unroll 2
    for (int nt = 0; nt < 16; nt++) {
        const int col = nt * 16 + lo;
        const float bm0 = sMW1[(hi ? 2 : 0) * HID + col];
        const float bm1 = hi ? 0.0f : sMW1[HID + col];
        v8f ch = {};
        ch = wmma_k4(ax0, ax1, bm0, bm1, ch);   // h_m again
        v8f cd = {};
        cd = wmma_k4(av0, av1, bm0, bm1, cd);   // dh = v @ M_W1 (no bias)
        const float bias = sMb1[col];
        const float4 w4 = *(const float4*)&sMW2[col * 4];
#pragma unroll
        for (int rr = 0; rr < 8; rr++) {
            const float t  = fast_tanh(ch[rr] + bias);
            const float dt = (1.0f - t * t) * cd[rr];
            dac[rr * 4 + 0] += dt * w4.x;
            dac[rr * 4 + 1] += dt * w4.y;
            dac[rr * 4 + 2] += dt * w4.z;
            dac[rr * 4 + 3] += dt * w4.w;
        }
    }
    xor_reduce16<32>(dac);

    // ---------------- Phase 4: assemble ddq and store ----------------
#pragma unroll
    for (int rr = 0; rr < 8; rr++) {
        const float a0 = aac[rr * 4 + 0], a1 = aac[rr * 4 + 1];
        const float a2 = aac[rr * 4 + 2], a3 = aac[rr * 4 + 3];
        const float d0 = dac[rr * 4 + 0], d1 = dac[rr * 4 + 1];
        const float d2 = dac[rr * 4 + 2], d3 = dac[rr * 4 + 3];
        const float dm00 = 2.0f * (d0 * a0 + d1 * a1);
        const float dm01 = d0 * a2 + d1 * a3 + a0 * d2 + a1 * d3;
        const float dm11 = 2.0f * (d2 * a2 + d3 * a3);
        const float ddq0 = m00a[rr] * dp0a[rr] + m01a[rr] * dp1a[rr]
                         + dm00 * p0a[rr] + dm01 * p1a[rr];
        const float ddq1 = m01a[rr] * dp0a[rr] + m11a[rr] * dp1a[rr]
                         + dm01 * p0a[rr] + dm11 * p1a[rr];
        if (lo == rr) {  // lanes rr and rr+16 store rows rr and rr+8
            const int gr = rowBase + rr + 8 * hi;
            float* o = out + (size_t)gr * 6;
            *(float2*)(o + 0) = make_float2(dxa[rr], vs1a[rr]);
            *(float2*)(o + 2) = make_float2(vs2a[rr], ddq0);
            *(float2*)(o + 4) = make_float2(ddq1, 0.0f);
        }
    }
}

extern "C" void kernel_launch(void* const* d_in, const int* in_sizes, int n_in,
                              void* d_out, int out_size, void* d_ws, size_t ws_size,
                              hipStream_t stream) {
    (void)n_in; (void)d_ws; (void)ws_size; (void)out_size;
    // setup_inputs order: t, y, M_W1, M_b1, M_W2, M_b2, H_W1, H_b1, H_W2, H_b2,
    //                     g_W1, g_b1, g_W2, g_b2
    const float* y   = (const float*)d_in[1];
    const float* MW1 = (const float*)d_in[2];
    const float* Mb1 = (const float*)d_in[3];
    const float* MW2 = (const float*)d_in[4];
    const float* Mb2 = (const float*)d_in[5];
    const float* HW1 = (const float*)d_in[6];
    const float* Hb1 = (const float*)d_in[7];
    const float* HW2 = (const float*)d_in[8];
    const float* GW1 = (const float*)d_in[10];
    const float* Gb1 = (const float*)d_in[11];
    const float* GW2 = (const float*)d_in[12];
    const float* Gb2 = (const float*)d_in[13];
    float* out = (float*)d_out;

    const int bs     = in_sizes[1] / 6;        // 131072
    const int tiles  = bs / 16;                // 8192 (16 rows per wave)
    const int blocks = tiles / 8;              // 8 waves per 256-thread block

    hnn_fused_kernel<<<blocks, 256, 0, stream>>>(y, out, MW1, Mb1, MW2, Mb2,
                                                 HW1, Hb1, HW2, GW1, Gb1, GW2, Gb2);
}